// BandSplit_40553081208888
// MI455X (gfx1250) — compile-verified
//
#include <hip/hip_runtime.h>
#include <math.h>

#define NBANDS 62
#define DIM    384
#define XW     4100   // sum of band widths
#define NTOK   4096   // B*T

typedef __attribute__((ext_vector_type(2))) float v2f;
typedef __attribute__((ext_vector_type(8))) float v8f;

struct Params {
    const float* x;
    float*       out;
    const float* W[NBANDS];   // (w, 384) row-major
    const float* G[NBANDS];   // (w,)
    const float* Bv[NBANDS];  // (384,)
    int off[NBANDS];
    int bw[NBANDS];
};

__launch_bounds__(256)
__global__ void bandsplit_rmsnorm_linear_wmma(Params p) {
    // Max band width 516; stride padded by +4 floats to avoid LDS bank conflicts
    __shared__ float a_sh[16 * 520];
    __shared__ float red[256];
    __shared__ float s_sh[16];

    const int band = blockIdx.y;
    const int tile = blockIdx.x;
    const int w    = p.bw[band];       // multiple of 4
    const int off  = p.off[band];
    const int SAS  = w + 4;            // even stride -> float2-aligned LDS reads
    const float* Wb = p.W[band];
    const float* Gb = p.G[band];
    const float* Bb = p.Bv[band];

    const int tid = threadIdx.x;

    // ---- Phase 1: stage A = x * gamma into LDS; per-token sum of squares ----
    // x is streamed exactly once across the whole grid -> non-temporal loads,
    // keep L2 capacity for the weight matrices (reused by all 256 token tiles).
    const int m   = tid >> 4;          // token row 0..15
    const int j16 = tid & 15;
    const size_t tok0 = (size_t)tile * 16;
    const float* xrow = p.x + (tok0 + (size_t)m) * (size_t)XW + off;

    float ss = 0.f;
    for (int k = j16; k < w; k += 16) {
        float v = __builtin_nontemporal_load(&xrow[k]);
        ss += v * v;
        a_sh[m * SAS + k] = v * Gb[k];
    }
    red[tid] = ss;
    __syncthreads();
    if (tid < 16) {
        float t = 0.f;
        #pragma unroll
        for (int i = 0; i < 16; ++i) t += red[tid * 16 + i];
        s_sh[tid] = sqrtf((float)w) * rsqrtf(t);   // sqrt(d) / ||x||
    }
    __syncthreads();

    // ---- Phase 2: WMMA over K, 8 waves x 3 N-tiles = 384 columns ----
    const int wave = tid >> 5;
    const int lane = tid & 31;
    const int c    = lane & 15;        // A: row M; B/D: column N
    const int hi   = lane >> 4;        // half-wave select
    const int koff = hi * 2;           // A/B K-pair per half-wave

    // Row scales for D layout: VGPR j holds row M = j + 8*hi
    float sv[8];
    #pragma unroll
    for (int j = 0; j < 8; ++j) sv[j] = s_sh[hi * 8 + j];

    for (int i = 0; i < 3; ++i) {
        const int n0 = (wave * 3 + i) * 16;
        const float bn = Bb[n0 + c];
        v8f acc = {};
        for (int k0 = 0; k0 < w; k0 += 4) {
            // A fragment: lanes 0-15 K={k0,k0+1}, lanes 16-31 K={k0+2,k0+3}, M=c
            v2f a = *(const v2f*)&a_sh[c * SAS + k0 + koff];
            // B fragment: row K, col n0+c; rows k0+koff and k0+koff+1
            // Default (RT) temporal hint: weights are L2-resident and reused.
            v2f b;
            b.x = Wb[(size_t)(k0 + koff)     * DIM + n0 + c];
            b.y = Wb[(size_t)(k0 + koff + 1) * DIM + n0 + c];
            // D = A*B + C  (fp32 WMMA, full reference precision)
            acc = __builtin_amdgcn_wmma_f32_16x16x4_f32(
                false, a, false, b, (short)0, acc, false, false);
        }
        // ---- Epilogue: scale rows by s, add bias, stream out (B,T,62,384) ----
        // Output (390 MB) is write-once -> non-temporal stores, don't churn L2.
        #pragma unroll
        for (int j = 0; j < 8; ++j) {
            const int r = hi * 8 + j;
            const float val = acc[j] * sv[j] + bn;
            const size_t oidx = (tok0 + (size_t)r) * (size_t)(NBANDS * DIM)
                              + (size_t)band * DIM + (size_t)(n0 + c);
            __builtin_nontemporal_store(val, &p.out[oidx]);
        }
    }
}

extern "C" void kernel_launch(void* const* d_in, const int* in_sizes, int n_in,
                              void* d_out, int out_size, void* d_ws, size_t ws_size,
                              hipStream_t stream) {
    static const int bands[NBANDS] = {
        8,8,8,8,8,8,8,8,8,8,8,8,8,8,8,8,8,8,8,8,8,8,8,8,
        16,16,16,16,16,16,16,16,16,16,16,16,
        48,48,48,48,48,48,48,48,
        96,96,96,96,96,96,96,96,
        192,192,192,192,192,192,192,192,
        512, 516
    };

    Params p;
    p.x   = (const float*)d_in[0];
    p.out = (float*)d_out;
    int off = 0;
    for (int i = 0; i < NBANDS; ++i) {
        p.G[i]  = (const float*)d_in[1 + i];
        p.W[i]  = (const float*)d_in[1 + NBANDS + i];
        p.Bv[i] = (const float*)d_in[1 + 2 * NBANDS + i];
        p.off[i] = off;
        p.bw[i]  = bands[i];
        off += bands[i];
    }

    dim3 grid(NTOK / 16, NBANDS);  // 256 token tiles x 62 bands
    bandsplit_rmsnorm_linear_wmma<<<grid, 256, 0, stream>>>(p);
}